// GATLayer_67611375174004
// MI455X (gfx1250) — compile-verified
//
#include <hip/hip_runtime.h>
#include <math.h>

#define HEADS 4
#define CPH 16            // channels per head
#define DIM 64            // HEADS*CPH == IN_DIM == OUT_DIM
#define NEG_SLOPE 0.2f
#define SOFTMAX_EPS 1e-16f
#define LN_EPS 1e-5f

typedef __attribute__((ext_vector_type(2))) float v2f;
typedef __attribute__((ext_vector_type(8))) float v8f;

// ---- order-preserving float <-> uint map for atomic max on floats ----
__device__ __forceinline__ unsigned encF(float f) {
  unsigned u = __float_as_uint(f);
  return (u & 0x80000000u) ? ~u : (u | 0x80000000u);
}
__device__ __forceinline__ float decF(unsigned u) {
  return __uint_as_float((u & 0x80000000u) ? (u & 0x7fffffffu) : ~u);
}
// encF(-inf) == ~0xFF800000 == 0x007FFFFF
#define ENC_NEG_INF 0x007FFFFFu

__device__ __forceinline__ float lrelu(float v) {
  return v >= 0.0f ? v : NEG_SLOPE * v;
}

// ---------------- init: zero accumulators, seed max cells ----------------
__global__ void gat_init(float* __restrict__ out, unsigned* __restrict__ maxu,
                         float* __restrict__ denom, int n) {
  int i = blockIdx.x * blockDim.x + threadIdx.x;
  if (i < n * DIM) out[i] = 0.0f;
  if (i < n * HEADS) { maxu[i] = ENC_NEG_INF; denom[i] = 0.0f; }
}

// ---------------- h = x @ W via V_WMMA_F32_16X16X4_F32 ----------------
// 1 wave per 16x16 output tile; 4 waves/block cover the 4 head column-tiles.
__global__ void gat_gemm_wmma(const float* __restrict__ x, const float* __restrict__ W,
                              float* __restrict__ h, int n) {
  const int lane  = threadIdx.x & 31;
  const int wave  = threadIdx.x >> 5;        // 0..3 : column tile (== head)
  const int row0  = blockIdx.x * 16;
  const int col0  = wave * 16;
  const int m     = lane & 15;               // M (A) / N (B,C,D) index
  const int kHalf = (lane >> 4) * 2;         // lanes 0-15: K0,K1 ; lanes 16-31: K2,K3

  const int rowA = min(row0 + m, n - 1);     // clamped gather row (tail-safe)

  v8f acc = {};
  #pragma unroll
  for (int k0 = 0; k0 < DIM; k0 += 4) {
    v2f a, b;
    a.x = x[(size_t)rowA * DIM + k0 + kHalf + 0];
    a.y = x[(size_t)rowA * DIM + k0 + kHalf + 1];
    b.x = W[(size_t)(k0 + kHalf + 0) * DIM + col0 + m];
    b.y = W[(size_t)(k0 + kHalf + 1) * DIM + col0 + m];
    // D = A(16x4) * B(4x16) + C   -> v_wmma_f32_16x16x4_f32
    acc = __builtin_amdgcn_wmma_f32_16x16x4_f32(
        /*neg_a=*/false, a, /*neg_b=*/false, b,
        /*c_mod=*/(short)0, acc, /*reuse_a=*/false, /*reuse_b=*/false);
  }

  // C/D layout: VGPR r -> M = r + (lane<16 ? 0 : 8), N = lane&15
  const int mOut = (lane >> 4) * 8;
  #pragma unroll
  for (int r = 0; r < 8; ++r) {
    int row = row0 + mOut + r;
    if (row < n) h[(size_t)row * DIM + col0 + m] = acc[r];
  }
}

// ---------------- per-(node,head) attention dots ----------------
__global__ void gat_att(const float* __restrict__ h, const float* __restrict__ att_src,
                        const float* __restrict__ att_dst, float* __restrict__ a_src,
                        float* __restrict__ a_dst, int n) {
  int i = blockIdx.x * blockDim.x + threadIdx.x;   // i = node*HEADS + head
  if (i >= n * HEADS) return;
  int node = i >> 2, hd = i & 3;
  const float* hp = h + (size_t)node * DIM + hd * CPH;
  const float* as = att_src + hd * CPH;
  const float* ad = att_dst + hd * CPH;
  float s = 0.0f, d = 0.0f;
  #pragma unroll
  for (int c = 0; c < CPH; ++c) { float v = hp[c]; s += v * as[c]; d += v * ad[c]; }
  a_src[i] = s; a_dst[i] = d;
}

// ---------------- pass 1: segment max over incoming edges ----------------
__global__ void gat_edge_max(const int* __restrict__ ei, const float* __restrict__ a_src,
                             const float* __restrict__ a_dst, unsigned* __restrict__ maxu,
                             int e_cnt) {
  int i = blockIdx.x * blockDim.x + threadIdx.x;   // i = edge*HEADS + head
  if (i >= e_cnt * HEADS) return;
  int e = i >> 2, hd = i & 3;
  int s = ei[e], d = ei[e_cnt + e];
  float v = lrelu(a_src[s * HEADS + hd] + a_dst[d * HEADS + hd]);
  atomicMax(&maxu[d * HEADS + hd], encF(v));
}

// ---------------- pass 2: softmax denominator ----------------
__global__ void gat_edge_sum(const int* __restrict__ ei, const float* __restrict__ a_src,
                             const float* __restrict__ a_dst, const unsigned* __restrict__ maxu,
                             float* __restrict__ denom, int e_cnt) {
  int i = blockIdx.x * blockDim.x + threadIdx.x;
  if (i >= e_cnt * HEADS) return;
  int e = i >> 2, hd = i & 3;
  int s = ei[e], d = ei[e_cnt + e];
  float v = lrelu(a_src[s * HEADS + hd] + a_dst[d * HEADS + hd]);
  float m = decF(maxu[d * HEADS + hd]);
  atomicAdd(&denom[d * HEADS + hd], expf(v - m));
}

// ---------------- pass 3: alpha-weighted scatter of h[src] ----------------
// one wave (32 lanes) per edge; each lane handles channels lane and lane+32
__global__ void gat_edge_scatter(const int* __restrict__ ei, const float* __restrict__ a_src,
                                 const float* __restrict__ a_dst, const unsigned* __restrict__ maxu,
                                 const float* __restrict__ denom, const float* __restrict__ h,
                                 float* __restrict__ out, int e_cnt) {
  int lane = threadIdx.x & 31;
  int e = blockIdx.x * (blockDim.x >> 5) + (threadIdx.x >> 5);
  if (e >= e_cnt) return;
  int s = ei[e], d = ei[e_cnt + e];
  #pragma unroll
  for (int half = 0; half < 2; ++half) {
    int ch = lane + 32 * half;
    int hd = ch >> 4;                 // head = channel / 16
    float v = lrelu(a_src[s * HEADS + hd] + a_dst[d * HEADS + hd]);
    float m = decF(maxu[d * HEADS + hd]);
    float alpha = expf(v - m) / (denom[d * HEADS + hd] + SOFTMAX_EPS);
    atomicAdd(&out[(size_t)d * DIM + ch], alpha * h[(size_t)s * DIM + ch]);
  }
}

// ---------------- bias + LayerNorm, wave per node (wave32 shfl) ----------------
__global__ void gat_layernorm(float* __restrict__ out, const float* __restrict__ bias,
                              const float* __restrict__ gamma, const float* __restrict__ beta,
                              int n) {
  int lane = threadIdx.x & 31;
  int node = blockIdx.x * (blockDim.x >> 5) + (threadIdx.x >> 5);
  if (node >= n) return;
  float v0 = out[(size_t)node * DIM + lane]      + bias[lane];
  float v1 = out[(size_t)node * DIM + lane + 32] + bias[lane + 32];
  float s = v0 + v1, sq = v0 * v0 + v1 * v1;
  #pragma unroll
  for (int off = 16; off > 0; off >>= 1) {
    s  += __shfl_xor(s,  off, 32);
    sq += __shfl_xor(sq, off, 32);
  }
  float mu  = s * (1.0f / DIM);
  float var = sq * (1.0f / DIM) - mu * mu;
  float inv = rsqrtf(var + LN_EPS);
  out[(size_t)node * DIM + lane]      = (v0 - mu) * inv * gamma[lane]      + beta[lane];
  out[(size_t)node * DIM + lane + 32] = (v1 - mu) * inv * gamma[lane + 32] + beta[lane + 32];
}

extern "C" void kernel_launch(void* const* d_in, const int* in_sizes, int n_in,
                              void* d_out, int out_size, void* d_ws, size_t ws_size,
                              hipStream_t stream) {
  const float* x       = (const float*)d_in[0];   // [N,64]
  const int*   ei      = (const int*)d_in[1];     // [2,E]
  const float* W       = (const float*)d_in[2];   // [64,64]
  const float* att_src = (const float*)d_in[3];   // [4,16]
  const float* att_dst = (const float*)d_in[4];   // [4,16]
  const float* bias    = (const float*)d_in[5];   // [64]
  const float* gamma   = (const float*)d_in[6];   // [64]
  const float* beta    = (const float*)d_in[7];   // [64]
  float* out = (float*)d_out;

  const int n = in_sizes[0] / DIM;      // 100000
  const int e = in_sizes[1] / 2;        // 1000000

  // workspace carve-up
  float*    h     = (float*)d_ws;                    // n*64 f32
  float*    a_src = h + (size_t)n * DIM;             // n*4
  float*    a_dst = a_src + (size_t)n * HEADS;       // n*4
  unsigned* maxu  = (unsigned*)(a_dst + (size_t)n * HEADS); // n*4
  float*    denom = (float*)(maxu + (size_t)n * HEADS);     // n*4

  const int T = 256;

  gat_init<<<(n * DIM + T - 1) / T, T, 0, stream>>>(out, maxu, denom, n);

  // 16-row tiles, 4 waves/block each doing one 16-col (head) tile
  gat_gemm_wmma<<<(n + 15) / 16, 128, 0, stream>>>(x, W, h, n);

  gat_att<<<(n * HEADS + T - 1) / T, T, 0, stream>>>(h, att_src, att_dst, a_src, a_dst, n);

  gat_edge_max<<<((size_t)e * HEADS + T - 1) / T, T, 0, stream>>>(ei, a_src, a_dst, maxu, e);

  gat_edge_sum<<<((size_t)e * HEADS + T - 1) / T, T, 0, stream>>>(ei, a_src, a_dst, maxu, denom, e);

  // one wave per edge (8 waves per 256-thread block)
  gat_edge_scatter<<<(e + 7) / 8, T, 0, stream>>>(ei, a_src, a_dst, maxu, denom, h, out, e);

  // one wave per node
  gat_layernorm<<<(n + 7) / 8, T, 0, stream>>>(out, bias, gamma, beta, n);
}